// SinkhornDistance_9543417332491
// MI455X (gfx1250) — compile-verified
//
#include <hip/hip_runtime.h>
#include <hip/hip_bf16.h>
#include <math.h>

// Problem constants (match reference)
#define BATCH 8
#define NN    2048
#define DD    64
#define EPSF      0.01f
#define INV_EPS   100.0f
#define THRESH    0.1f
#define MAX_ITER  100

typedef __attribute__((ext_vector_type(2))) float v2f;
typedef __attribute__((ext_vector_type(8))) float v8f;

// log(1/2048 + 1e-8) — constant-folded by the compiler
__device__ __forceinline__ float log_mu() { return logf(1.0f / 2048.0f + 1e-8f); }

// ---------------------------------------------------------------------------
// Init: zero u, v, err, done, and the output accumulator.
// ---------------------------------------------------------------------------
__global__ void init_state(float* __restrict__ u, float* __restrict__ v,
                           float* __restrict__ err, int* __restrict__ done,
                           float* __restrict__ out, int n) {
    int idx = blockIdx.x * blockDim.x + threadIdx.x;
    if (idx < n) { u[idx] = 0.0f; v[idx] = 0.0f; }
    if (idx == 0) { *err = 0.0f; *done = 0; *out = 0.0f; }
}

// ---------------------------------------------------------------------------
// Row squared norms: one wave (32 lanes) per row of 64 floats.
// ---------------------------------------------------------------------------
__global__ void row_norms(const float* __restrict__ p, float* __restrict__ out,
                          int rows) {
    int wave = (blockIdx.x * blockDim.x + threadIdx.x) >> 5;
    int lane = threadIdx.x & 31;
    if (wave >= rows) return;
    const float* r = p + (size_t)wave * DD;
    float a = r[lane];
    float b = r[lane + 32];
    float s = a * a + b * b;
    #pragma unroll
    for (int off = 16; off > 0; off >>= 1) s += __shfl_xor(s, off, 32);
    if (lane == 0) out[wave] = s;
}

// ---------------------------------------------------------------------------
// Cost matrix: C[b,i,j] = xn[i] + yn[j] - 2 * (X Y^T)[i,j] via f32 WMMA.
// One wave computes a 32x32 tile (2x2 WMMA 16x16 tiles), K=64 in steps of 4.
// A frag (16x4 f32): lanes 0-15 hold (M=lane, K=0,1); lanes 16-31 (M, K=2,3).
// B frag (4x16 f32): symmetric over N -> loads Y row-major like A loads X.
// C/D (16x16 f32, 8 VGPRs): row = r + 8*half, col = lane&15.
// ---------------------------------------------------------------------------
__global__ void cost_gemm(const float* __restrict__ x, const float* __restrict__ y,
                          const float* __restrict__ xn, const float* __restrict__ yn,
                          float* __restrict__ C) {
    int wave = (blockIdx.x * blockDim.x + threadIdx.x) >> 5;
    int lane = threadIdx.x & 31;
    int b  = wave >> 12;            // 64*64 = 4096 tiles per batch
    int t  = wave & 4095;
    int i0 = (t >> 6) << 5;
    int j0 = (t & 63) << 5;

    const float* xb = x + (size_t)b * NN * DD;
    const float* yb = y + (size_t)b * NN * DD;

    int m    = lane & 15;
    int half = lane >> 4;

    int rowA0 = i0 + m, rowA1 = rowA0 + 16;
    int rowB0 = j0 + m, rowB1 = rowB0 + 16;

    v8f acc00 = {}, acc01 = {}, acc10 = {}, acc11 = {};

    #pragma unroll
    for (int k = 0; k < DD; k += 4) {
        int kk = k + half * 2;
        v2f a0 = *(const v2f*)(xb + (size_t)rowA0 * DD + kk);
        v2f a1 = *(const v2f*)(xb + (size_t)rowA1 * DD + kk);
        v2f b0 = *(const v2f*)(yb + (size_t)rowB0 * DD + kk);
        v2f b1 = *(const v2f*)(yb + (size_t)rowB1 * DD + kk);
        acc00 = __builtin_amdgcn_wmma_f32_16x16x4_f32(false, a0, false, b0,
                                                      (short)0, acc00, false, false);
        acc01 = __builtin_amdgcn_wmma_f32_16x16x4_f32(false, a0, false, b1,
                                                      (short)0, acc01, false, false);
        acc10 = __builtin_amdgcn_wmma_f32_16x16x4_f32(false, a1, false, b0,
                                                      (short)0, acc10, false, false);
        acc11 = __builtin_amdgcn_wmma_f32_16x16x4_f32(false, a1, false, b1,
                                                      (short)0, acc11, false, false);
    }

    const float* xnb = xn + (size_t)b * NN;
    const float* ynb = yn + (size_t)b * NN;
    float yn0 = ynb[j0 + m];
    float yn1 = ynb[j0 + 16 + m];
    float* Cb = C + (size_t)b * NN * NN;

    #pragma unroll
    for (int r = 0; r < 8; ++r) {
        int ra = i0 + half * 8 + r;   // rows of acc00/acc01
        int rb = ra + 16;             // rows of acc10/acc11
        int ca = j0 + m;
        int cb = ca + 16;
        float xa  = xnb[ra];
        float xbn = xnb[rb];
        Cb[(size_t)ra * NN + ca] = xa  + yn0 - 2.0f * acc00[r];
        Cb[(size_t)ra * NN + cb] = xa  + yn1 - 2.0f * acc01[r];
        Cb[(size_t)rb * NN + ca] = xbn + yn0 - 2.0f * acc10[r];
        Cb[(size_t)rb * NN + cb] = xbn + yn1 - 2.0f * acc11[r];
    }
}

// ---------------------------------------------------------------------------
// u update: one 256-thread block per (b,i). Streams the row with b128 loads.
// u_new = eps*(log_mu - log(sum_j exp((-C+u_i+v_j)/eps) + 1e-6)) + u_i
// Accumulates sum |u_new - u| into err. No-op once converged (done flag).
// ---------------------------------------------------------------------------
__global__ void row_update(const float* __restrict__ C, float* __restrict__ u,
                           const float* __restrict__ v, float* __restrict__ err,
                           const int* __restrict__ done) {
    if (*done) return;
    int blk = blockIdx.x;
    int b = blk >> 11;
    int i = blk & (NN - 1);
    const float4* Crow = (const float4*)(C + ((size_t)b * NN + i) * NN);
    const float4* vb   = (const float4*)(v + (size_t)b * NN);
    float ui = u[(size_t)b * NN + i];

    float s = 0.0f;
    #pragma unroll
    for (int j = threadIdx.x; j < NN / 4; j += 256) {
        float4 c  = Crow[j];
        float4 vv = vb[j];
        s += expf((ui + vv.x - c.x) * INV_EPS);
        s += expf((ui + vv.y - c.y) * INV_EPS);
        s += expf((ui + vv.z - c.z) * INV_EPS);
        s += expf((ui + vv.w - c.w) * INV_EPS);
    }

    #pragma unroll
    for (int off = 16; off > 0; off >>= 1) s += __shfl_xor(s, off, 32);
    __shared__ float red[8];
    int lane = threadIdx.x & 31, wv = threadIdx.x >> 5;
    if (lane == 0) red[wv] = s;
    __syncthreads();
    if (threadIdx.x == 0) {
        float tot = 0.0f;
        #pragma unroll
        for (int w = 0; w < 8; ++w) tot += red[w];
        float lse = logf(tot + 1e-6f);
        float un  = EPSF * (log_mu() - lse) + ui;
        u[(size_t)b * NN + i] = un;
        atomicAdd(err, fabsf(un - ui));
    }
}

// ---------------------------------------------------------------------------
// v update: 1024-thread block = 32 waves. Each wave covers 64 consecutive
// columns (float2 per lane -> coalesced 256B global_load_b64 per row step)
// over a 64-row i-slice. LDS reduction across waves, 64 threads finalize.
// ---------------------------------------------------------------------------
__global__ void __launch_bounds__(1024)
col_update(const float* __restrict__ C, const float* __restrict__ u,
           float* __restrict__ v, const int* __restrict__ done) {
    if (*done) return;
    int blk = blockIdx.x;          // BATCH * 32 blocks
    int b  = blk >> 5;
    int j0 = (blk & 31) << 6;      // 64 columns per block
    int lane = threadIdx.x & 31;
    int wv   = threadIdx.x >> 5;   // 0..31
    int col  = j0 + lane * 2;

    const float* ub = u + (size_t)b * NN;
    const float* Cb = C + (size_t)b * NN * NN;
    float2 vj = *(const float2*)(v + (size_t)b * NN + col);

    float2 s = make_float2(0.0f, 0.0f);
    int ibase = wv << 6;           // 64 rows per wave
    for (int ii = 0; ii < 64; ++ii) {
        int i = ibase + ii;
        float uiv = ub[i];
        float2 c = *(const float2*)(Cb + (size_t)i * NN + col);
        s.x += expf((uiv + vj.x - c.x) * INV_EPS);
        s.y += expf((uiv + vj.y - c.y) * INV_EPS);
    }

    __shared__ float part[32 * 64];
    *(float2*)&part[wv * 64 + lane * 2] = s;
    __syncthreads();
    if (threadIdx.x < 64) {
        int t = threadIdx.x;
        float tot = 0.0f;
        #pragma unroll
        for (int w = 0; w < 32; ++w) tot += part[w * 64 + t];
        float lse = logf(tot + 1e-6f);
        float vold = v[(size_t)b * NN + j0 + t];
        v[(size_t)b * NN + j0 + t] = EPSF * (log_mu() - lse) + vold;
    }
}

// ---------------------------------------------------------------------------
// Convergence check: err = (sum |du|) / B ; freeze once err < THRESH.
// Also resets the err accumulator for the next iteration.
// ---------------------------------------------------------------------------
__global__ void conv_check(float* __restrict__ err, int* __restrict__ done) {
    if (!*done) {
        float e = *err * (1.0f / (float)BATCH);
        if (e < THRESH) *done = 1;
    }
    *err = 0.0f;
}

// ---------------------------------------------------------------------------
// Final cost: mean_b sum_ij exp(M) * C with b128 streaming, atomic into out.
// ---------------------------------------------------------------------------
__global__ void final_cost(const float* __restrict__ C, const float* __restrict__ u,
                           const float* __restrict__ v, float* __restrict__ out) {
    int blk = blockIdx.x;
    int b = blk >> 11;
    int i = blk & (NN - 1);
    const float4* Crow = (const float4*)(C + ((size_t)b * NN + i) * NN);
    const float4* vb   = (const float4*)(v + (size_t)b * NN);
    float ui = u[(size_t)b * NN + i];

    float s = 0.0f;
    #pragma unroll
    for (int j = threadIdx.x; j < NN / 4; j += 256) {
        float4 c  = Crow[j];
        float4 vv = vb[j];
        s += expf((ui + vv.x - c.x) * INV_EPS) * c.x;
        s += expf((ui + vv.y - c.y) * INV_EPS) * c.y;
        s += expf((ui + vv.z - c.z) * INV_EPS) * c.z;
        s += expf((ui + vv.w - c.w) * INV_EPS) * c.w;
    }
    #pragma unroll
    for (int off = 16; off > 0; off >>= 1) s += __shfl_xor(s, off, 32);
    __shared__ float red[8];
    int lane = threadIdx.x & 31, wv = threadIdx.x >> 5;
    if (lane == 0) red[wv] = s;
    __syncthreads();
    if (threadIdx.x == 0) {
        float tot = 0.0f;
        #pragma unroll
        for (int w = 0; w < 8; ++w) tot += red[w];
        atomicAdd(out, tot * (1.0f / (float)BATCH));
    }
}

// ---------------------------------------------------------------------------
extern "C" void kernel_launch(void* const* d_in, const int* in_sizes, int n_in,
                              void* d_out, int out_size, void* d_ws, size_t ws_size,
                              hipStream_t stream) {
    const float* x = (const float*)d_in[0];   // [B, N, D] f32
    const float* y = (const float*)d_in[1];   // [B, N, D] f32
    float* out = (float*)d_out;               // scalar f32

    // Workspace layout
    char* ws = (char*)d_ws;
    float* C   = (float*)ws;                                 // 8*2048*2048*4 = 128 MB
    float* xn  = (float*)(ws + (size_t)BATCH * NN * NN * 4); // B*N
    float* yn  = xn + BATCH * NN;
    float* u   = yn + BATCH * NN;
    float* v   = u  + BATCH * NN;
    float* err = v  + BATCH * NN;
    int*  done = (int*)(err + 1);

    const int rows = BATCH * NN;  // 16384

    init_state<<<(rows + 255) / 256, 256, 0, stream>>>(u, v, err, done, out, rows);
    row_norms<<<rows / 8, 256, 0, stream>>>(x, xn, rows);
    row_norms<<<rows / 8, 256, 0, stream>>>(y, yn, rows);

    // 8 batches * 64*64 tiles, 8 waves per 256-thread block
    cost_gemm<<<(BATCH * 64 * 64) / 8, 256, 0, stream>>>(x, y, xn, yn, C);

    for (int it = 0; it < MAX_ITER; ++it) {
        row_update<<<rows, 256, 0, stream>>>(C, u, v, err, done);
        col_update<<<BATCH * (NN / 64), 1024, 0, stream>>>(C, u, v, done);
        conv_check<<<1, 1, 0, stream>>>(err, done);
    }

    final_cost<<<rows, 256, 0, stream>>>(C, u, v, out);
}